// PseudoChiSquareLoss_29807073034374
// MI455X (gfx1250) — compile-verified
//
#include <hip/hip_runtime.h>

#define NBINS_MAX 128
#define REPLICAS  8
#define TPB       256

typedef __attribute__((ext_vector_type(2))) float v2f;
typedef __attribute__((ext_vector_type(4))) float fv4;
typedef __attribute__((ext_vector_type(8))) float v8f;

// ---------------------------------------------------------------------------
// Kernel 0: zero the global accumulators living in d_ws (harness poisons it).
// ---------------------------------------------------------------------------
__global__ void zero_ws_kernel(float* __restrict__ g, int n) {
    int i = blockIdx.x * blockDim.x + threadIdx.x;
    if (i < n) g[i] = 0.0f;
}

// ---------------------------------------------------------------------------
// Kernel 1: streaming interpolated-histogram pass. Bandwidth-bound:
// float4 non-temporal loads, bin index computed arithmetically (reference's
// delta = (bt[-1]-bt[0])/nbins with a uniform table), LDS ds_add_f32 atomics
// into per-wave replicated sub-histograms, one global flush per block.
// Layout: repl[2*b] = hist[b], repl[2*b+1] = wss[b].
// ---------------------------------------------------------------------------
__device__ __forceinline__ void accum_one(float ov, float wv, float b0,
                                          float inv_delta, int nbins,
                                          float* __restrict__ repl) {
    float t    = (ov - b0) * inv_delta;
    float fj   = floorf(t);
    int   j    = (int)fj;
    float frac = t - fj;
    float ws   = frac * wv;        // (obs - left)/delta * w  -> bin j+1
    float wp   = wv - ws;          // (right - obs)/delta * w -> bin j
    if (j >= 0 && j <= nbins - 3) {
        float* p = repl + 2 * (j + 1);
        atomicAdd(p,     ws);
        atomicAdd(p + 1, ws * ws);
    }
    if (j >= 1 && j <= nbins - 2) {
        float* p = repl + 2 * j;
        atomicAdd(p,     wp);
        atomicAdd(p + 1, wp * wp);
    }
}

__global__ __launch_bounds__(TPB) void hist_kernel(
    const float* __restrict__ obs, const float* __restrict__ wgt,
    const float* __restrict__ bins, float* __restrict__ gacc,
    int n, int nbins) {
    __shared__ float lh[REPLICAS][2 * NBINS_MAX];
    for (int i = threadIdx.x; i < REPLICAS * 2 * NBINS_MAX; i += TPB)
        ((float*)lh)[i] = 0.0f;
    __syncthreads();

    const float b0        = bins[0];
    const float inv_delta = (float)nbins / (bins[nbins] - b0);
    float* repl = lh[(threadIdx.x >> 5) & (REPLICAS - 1)];

    const int gid  = blockIdx.x * TPB + threadIdx.x;
    const int gsz  = gridDim.x * TPB;
    const int nvec = n >> 2;
    const fv4* __restrict__ obs4 = (const fv4*)obs;
    const fv4* __restrict__ wgt4 = (const fv4*)wgt;

    for (int i = gid; i < nvec; i += gsz) {
        fv4 o = __builtin_nontemporal_load(&obs4[i]);
        fv4 w = __builtin_nontemporal_load(&wgt4[i]);
#pragma unroll
        for (int k = 0; k < 4; ++k)
            accum_one(o[k], w[k], b0, inv_delta, nbins, repl);
    }
    // scalar tail
    for (int i = (nvec << 2) + gid; i < n; i += gsz)
        accum_one(obs[i], wgt[i], b0, inv_delta, nbins, repl);

    __syncthreads();
    for (int i = threadIdx.x; i < 2 * nbins; i += TPB) {
        float s = 0.0f;
#pragma unroll
        for (int r = 0; r < REPLICAS; ++r) s += lh[r][i];
        if (s != 0.0f) atomicAdd(&gacc[i], s);
    }
}

// ---------------------------------------------------------------------------
// Wave32 reduction via V_WMMA_F32_16X16X4_F32. A = all-ones 16x4, B holds the
// 32 lane partials (second B VGPR = 0). D[m][n] = sum_k B[k][n], so every row
// of D is the vector of 16 column sums; D's first VGPR is [cs0..cs15,
// cs0..cs15] across lanes. Four shuffle-xors finish the sum; exact f32 math.
// ---------------------------------------------------------------------------
__device__ __forceinline__ float wave_sum32(float x) {
    v2f a; a[0] = 1.0f; a[1] = 1.0f;
    v2f b; b[0] = x;    b[1] = 0.0f;
    v8f c = {};
    v8f d = __builtin_amdgcn_wmma_f32_16x16x4_f32(
        /*neg_a=*/false, a, /*neg_b=*/false, b,
        /*c_mod=*/(short)0, c, /*reuse_a=*/false, /*reuse_b=*/false);
    float s = d[0];
    s += __shfl_xor(s, 1, 32);
    s += __shfl_xor(s, 2, 32);
    s += __shfl_xor(s, 4, 32);
    s += __shfl_xor(s, 8, 32);
    return s;  // total across all 32 lanes, in every lane
}

// ---------------------------------------------------------------------------
// Kernel 2: single-wave finalize. Computes sum_sim, sum_exp, then chi2.
// ---------------------------------------------------------------------------
__global__ __launch_bounds__(32) void finalize_kernel(
    const float* __restrict__ gacc, const float* __restrict__ histo_exp,
    float* __restrict__ out, int nbins) {
    const int l = threadIdx.x;  // 0..31, one full wave (EXEC all ones)
    float ph = 0.0f, pe = 0.0f;
    for (int b = l; b < nbins; b += 32) {
        ph += gacc[2 * b];
        pe += histo_exp[b];
    }
    const float sum_sim = wave_sum32(ph);
    const float sum_exp = wave_sum32(pe);

    const float EPS     = 1e-10f;
    const float inv_ss2 = 1.0f / ((sum_sim + EPS) * (sum_sim + EPS));
    const float inv_se2 = 1.0f / ((sum_exp + EPS) * (sum_exp + EPS));

    float pc = 0.0f;
    for (int b = l; b < nbins; b += 32) {
        float h   = gacc[2 * b];
        float wss = gacc[2 * b + 1];
        float he  = histo_exp[b];
        float us  = wss * inv_ss2 + EPS;
        float ue  = he * (1.0f - he / sum_exp) * inv_se2 + EPS;
        float dfn = h / sum_sim - he / sum_exp;
        pc += dfn * dfn / (us + ue);
    }
    const float chi2 = wave_sum32(pc);
    if (l == 0) out[0] = chi2;
}

// ---------------------------------------------------------------------------
extern "C" void kernel_launch(void* const* d_in, const int* in_sizes, int n_in,
                              void* d_out, int out_size, void* d_ws, size_t ws_size,
                              hipStream_t stream) {
    const float* obs  = (const float*)d_in[0];  // sim_observable
    // d_in[1] = exp_observable: unused in fixed-binning branch
    const float* wgt  = (const float*)d_in[2];  // weights
    const float* bins = (const float*)d_in[3];  // nbins+1 edges
    const float* hexp = (const float*)d_in[4];  // histo_exp
    const int n     = in_sizes[0];
    int       nbins = in_sizes[4];
    if (nbins > NBINS_MAX) nbins = NBINS_MAX;   // LDS capacity guard

    float* gacc = (float*)d_ws;                 // 2*nbins accumulators

    zero_ws_kernel<<<1, 256, 0, stream>>>(gacc, 2 * nbins);

    int nvec   = n >> 2;
    int blocks = (nvec + TPB - 1) / TPB;
    if (blocks > 1024) blocks = 1024;           // ~8K waves: saturates HBM
    if (blocks < 1)    blocks = 1;
    hist_kernel<<<blocks, TPB, 0, stream>>>(obs, wgt, bins, gacc, n, nbins);

    finalize_kernel<<<1, 32, 0, stream>>>(gacc, hexp, (float*)d_out, nbins);
}